// PointMLP_12206297055636
// MI455X (gfx1250) — compile-verified
//
#include <hip/hip_runtime.h>

// PointMLP-like pipeline for MI455X (gfx1250, wave32, WMMA, TDM).
// Stages: knn (2D) -> laplacian gather (LDS-staged) -> fp32 WMMA GEMM with
// TDM (tensor_load_to_lds) double-buffered tile staging
//         -> per-feature BN stats -> normalize+ReLU+residual (in-place on d_out).

#define B_ 4
#define N_ 2048
#define K_ 8
#define F_ 2048

typedef float v2f __attribute__((ext_vector_type(2)));
typedef float v4f __attribute__((ext_vector_type(4)));
typedef float v8f __attribute__((ext_vector_type(8)));
typedef unsigned int v4u __attribute__((ext_vector_type(4)));
typedef int v4i __attribute__((ext_vector_type(4)));
typedef int v8i __attribute__((ext_vector_type(8)));

// ---------------------------------------------------------------------------
// Stage 1: 2-D KNN (top-8 smallest pdist, stable ties -> lower index first)
// ---------------------------------------------------------------------------
__global__ __launch_bounds__(256) void knn_kernel(const float* __restrict__ xyz,
                                                  int* __restrict__ idx) {
  __shared__ float sx[N_];
  __shared__ float sy[N_];
  __shared__ float sxx[N_];
  const int b = blockIdx.x >> 3;
  const int nbase = (blockIdx.x & 7) << 8;
  const float* xb = xyz + b * N_ * 3;
  for (int i = threadIdx.x; i < N_; i += 256) {
    float x = xb[i * 3 + 0];
    float y = xb[i * 3 + 1];
    sx[i] = x; sy[i] = y; sxx[i] = x * x + y * y;
  }
  __syncthreads();

  const int n = nbase + threadIdx.x;
  const float xn = sx[n], yn = sy[n], xxn = sxx[n];

  float bd[K_];
  int   bi[K_];
#pragma unroll
  for (int j = 0; j < K_; ++j) { bd[j] = 3.4e38f; bi[j] = 0; }

  for (int m = 0; m < N_; ++m) {
    float inner = 2.0f * (xn * sx[m] + yn * sy[m]);
    float d = (xxn - inner) + sxx[m];      // matches reference expr; d(n,n)==0 exactly
    if (d < bd[K_ - 1]) {                  // strict <: later equal candidate loses (stable)
      bd[K_ - 1] = d; bi[K_ - 1] = m;
#pragma unroll
      for (int j = K_ - 1; j > 0; --j) {
        if (bd[j] < bd[j - 1]) {
          float td = bd[j]; bd[j] = bd[j - 1]; bd[j - 1] = td;
          int   ti = bi[j]; bi[j] = bi[j - 1]; bi[j - 1] = ti;
        }
      }
    }
  }
  int* op = idx + (b * N_ + n) * K_;
#pragma unroll
  for (int j = 0; j < K_; ++j) op[j] = bi[j];
}

// ---------------------------------------------------------------------------
// Stage 2: lap[b,f,n] = feat[b,f,n] - (1/8) sum_k feat[b, n, idx_flat[b*16384 + k*2048 + f]]
// 8 point-rows staged in 64KB LDS, XOR bank swizzle; fill loads batched so the
// 8 global_load_b128 issue before any ds_store (no per-load loadcnt stall).
// ---------------------------------------------------------------------------
__global__ __launch_bounds__(256) void lap_kernel(const float* __restrict__ feat,
                                                  const int* __restrict__ idx,
                                                  float* __restrict__ lap) {
  __shared__ float rows[8 * N_];                 // 64 KB
  const int b = blockIdx.x >> 8;
  const int nbase = (blockIdx.x & 255) << 3;
  const float* fb = feat + (size_t)b * F_ * N_;

#pragma unroll
  for (int half = 0; half < 2; ++half) {
    v4f tmp[8];
#pragma unroll
    for (int j = 0; j < 8; ++j) {
      int lin = threadIdx.x + (half * 8 + j) * 256;
      int row = lin >> 9;
      int c4 = (lin & 511) << 2;
      tmp[j] = *(const v4f*)(fb + (size_t)(nbase + row) * N_ + c4);
    }
#pragma unroll
    for (int j = 0; j < 8; ++j) {
      int lin = threadIdx.x + (half * 8 + j) * 256;
      int row = lin >> 9;
      int c4 = (lin & 511) << 2;
      *(v4f*)(rows + row * N_ + (c4 ^ (row << 3))) = tmp[j];  // swizzle preserves 16B blocks
    }
  }
  __syncthreads();

  const int nl  = threadIdx.x & 7;
  const int fl0 = threadIdx.x >> 3;
  const int swz = nl << 3;
  const float* ldsrow = rows + nl * N_;
  const int n = nbase + nl;
  const int* idxb = idx + b * (N_ * K_);

  for (int f = fl0; f < F_; f += 32) {
    float s = 0.0f;
#pragma unroll
    for (int k = 0; k < K_; ++k) {
      int c = idxb[k * 2048 + f];
      s += ldsrow[c ^ swz];
    }
    lap[((size_t)b * F_ + f) * N_ + n] = fb[(size_t)f * N_ + n] - s * 0.125f;
  }
}

// ---------------------------------------------------------------------------
// Stage 3: fp32 WMMA GEMM, tiles staged by the Tensor Data Mover.
// Block tile 128(f) x 64(m), BK=32, double-buffered LDS. The TDM descriptor's
// pad feature (pad after every 32 DWORDs by 4 DWORDs) produces the 36-float
// LDS row stride -> conflict-free float2 fragment reads.
// ---------------------------------------------------------------------------
#define BM 128
#define BN 64
#define BK 32
#define LDT 36

// D# builder for a 2D f32 tile: tile_dim0=32 cols, tile_dim1=rows,
// tensor row stride 2048 elements.  CDNA5 ISA.
// This toolchain's builtin takes 6 args (extra int32x8 group, zero-filled).
__device__ __forceinline__ void tdm_load_tile(unsigned lds_off,
                                              unsigned long long gaddr,
                                              int rows) {
  v4u g0;
  g0[0] = 1u;                                            // count=1, user mode
  g0[1] = lds_off;                                       // LDS byte address
  g0[2] = (unsigned)gaddr;                               // global_addr[31:0]
  g0[3] = (unsigned)((gaddr >> 32) & 0x01FFFFFFull)      // global_addr[56:32]
          | (2u << 30);                                  // type=2 ("image")
  v8i g1;
  g1[0] = (2 << 16)        // data_size = 4B
        | (1 << 20)        // pad_enable
        | (4 << 22)        // pad_interval: 2^(4+1)=32 DWORDs between pads
        | (3 << 25);       // pad_amount: 3+1 = 4 DWORDs  -> row stride 36 floats
  g1[1] = (int)((N_ & 0xFFFF) << 16);                    // tensor_dim0 lo16
  g1[2] = (int)(((unsigned)N_ >> 16) | ((N_ & 0xFFFF) << 16)); // dim0 hi / dim1 lo
  g1[3] = (int)(((unsigned)N_ >> 16) | (32u << 16));     // dim1 hi / tile_dim0=32
  g1[4] = rows & 0xFFFF;                                 // tile_dim1 ; tile_dim2=0
  g1[5] = N_;                                            // tensor_dim0_stride lo32
  g1[6] = 0;
  g1[7] = 0;
  v4i z4 = {0, 0, 0, 0};                                 // groups 2/3 unused (2D)
  v8i z8 = {0, 0, 0, 0, 0, 0, 0, 0};                     // extra group, zero-filled
  __builtin_amdgcn_tensor_load_to_lds(g0, g1, z4, z4, z8, 0);
}

__global__ __launch_bounds__(256) void gemm_kernel(const float* __restrict__ lap,
                                                   const float* __restrict__ W,
                                                   const float* __restrict__ b_lin,
                                                   float* __restrict__ trans) {
  __shared__ float As[2][BM * LDT];                      // 2 x 18 KB
  __shared__ float Bs[2][BN * LDT];                      // 2 x 9 KB
  const int mb  = blockIdx.x;
  const int fbk = blockIdx.y;
  const int b   = blockIdx.z;

  const float* A  = lap + (size_t)b * F_ * N_ + (size_t)fbk * BM * N_;
  const float* Bw = W + (size_t)mb * BN * N_;

  const int tid  = threadIdx.x;
  const int lane = tid & 31;
  const int w    = tid >> 5;
  const int wf   = (w & 3) * 32;
  const int wm   = (w >> 2) * 32;
  const int lrow = lane & 15;
  const int lk2  = (lane >> 4) << 1;

  // Opaque, never-taken store: TDM writes to LDS are invisible to the
  // compiler's alias analysis; this keeps it from undef-folding LDS reads.
  if (__builtin_expect(b_lin[0] > 1.0e30f, 0)) {
    As[0][0] = 1.0f; Bs[0][0] = 1.0f;
  }

  const unsigned asOff = (unsigned)(unsigned long long)(uintptr_t)&As[0][0];
  const unsigned bsOff = (unsigned)(unsigned long long)(uintptr_t)&Bs[0][0];
  const unsigned long long gA = (unsigned long long)(uintptr_t)A;
  const unsigned long long gB = (unsigned long long)(uintptr_t)Bw;
  const bool issuer = (tid < 32);                        // wave 0 drives the TDM

  if (issuer) {
    tdm_load_tile(asOff, gA, BM);
    tdm_load_tile(bsOff, gB, BN);
  }

  v8f acc00 = {}, acc01 = {}, acc10 = {}, acc11 = {};

  for (int it = 0; it < N_ / BK; ++it) {
    const int cur = it & 1;
    if (issuer) {
      if (it + 1 < N_ / BK) {
        unsigned long long koff = (unsigned long long)(it + 1) * (BK * 4);
        tdm_load_tile(asOff + (cur ^ 1) * (BM * LDT * 4), gA + koff, BM);
        tdm_load_tile(bsOff + (cur ^ 1) * (BN * LDT * 4), gB + koff, BN);
        __builtin_amdgcn_s_wait_tensorcnt(2);            // current pair landed
      } else {
        __builtin_amdgcn_s_wait_tensorcnt(0);
      }
    }
    __syncthreads();                                     // data visible to all waves

    const float* a_base = &As[cur][0];
    const float* b_base = &Bs[cur][0];
#pragma unroll
    for (int kk = 0; kk < BK; kk += 4) {
      v2f a0 = *(const v2f*)(a_base + (wf + lrow) * LDT + kk + lk2);
      v2f a1 = *(const v2f*)(a_base + (wf + 16 + lrow) * LDT + kk + lk2);
      v2f b0 = *(const v2f*)(b_base + (wm + lrow) * LDT + kk + lk2);
      v2f b1 = *(const v2f*)(b_base + (wm + 16 + lrow) * LDT + kk + lk2);
      acc00 = __builtin_amdgcn_wmma_f32_16x16x4_f32(false, a0, false, b0, (short)0, acc00, false, false);
      acc01 = __builtin_amdgcn_wmma_f32_16x16x4_f32(false, a0, false, b1, (short)0, acc01, false, false);
      acc10 = __builtin_amdgcn_wmma_f32_16x16x4_f32(false, a1, false, b0, (short)0, acc10, false, false);
      acc11 = __builtin_amdgcn_wmma_f32_16x16x4_f32(false, a1, false, b1, (short)0, acc11, false, false);
    }
    __syncthreads();                                     // done reading before overwrite
  }

  // D layout: VGPR v -> M = v + 8*(lane>>4), Nn = lane & 15
  const int Mhi = (lane >> 4) << 3;
  const int Nn  = lane & 15;
  float* out = trans + (size_t)b * F_ * N_;
  const int frow0 = fbk * BM + wf;
  const int mcol0 = mb * BN + wm;
  const float bl0 = b_lin[mcol0 + Nn];
  const float bl1 = b_lin[mcol0 + 16 + Nn];
#pragma unroll
  for (int v = 0; v < 8; ++v) {
    int M = v + Mhi;
    out[(size_t)(frow0 + M) * N_ + mcol0 + Nn]           = acc00[v] + bl0;
    out[(size_t)(frow0 + M) * N_ + mcol0 + 16 + Nn]      = acc01[v] + bl1;
    out[(size_t)(frow0 + 16 + M) * N_ + mcol0 + Nn]      = acc10[v] + bl0;
    out[(size_t)(frow0 + 16 + M) * N_ + mcol0 + 16 + Nn] = acc11[v] + bl1;
  }
}

// ---------------------------------------------------------------------------
// Stage 4: per-feature (axis 0,2) mean / rsqrt(var+eps). One block per f.
// ---------------------------------------------------------------------------
__global__ __launch_bounds__(256) void stats_kernel(const float* __restrict__ trans,
                                                    float* __restrict__ stats) {
  __shared__ float ss[256];
  __shared__ float sq[256];
  const int f = blockIdx.x;
  float s = 0.0f, q = 0.0f;
  for (int j = threadIdx.x; j < B_ * N_; j += 256) {
    int b = j >> 11;
    int m = j & (N_ - 1);
    float v = trans[((size_t)b * F_ + f) * N_ + m];
    s += v; q += v * v;
  }
  ss[threadIdx.x] = s; sq[threadIdx.x] = q;
  __syncthreads();
  for (int off = 128; off > 0; off >>= 1) {
    if (threadIdx.x < off) {
      ss[threadIdx.x] += ss[threadIdx.x + off];
      sq[threadIdx.x] += sq[threadIdx.x + off];
    }
    __syncthreads();
  }
  if (threadIdx.x == 0) {
    float mu  = ss[0] * (1.0f / 8192.0f);
    float var = sq[0] * (1.0f / 8192.0f) - mu * mu;
    stats[f]      = mu;
    stats[F_ + f] = rsqrtf(var + 1e-5f);
  }
}

// ---------------------------------------------------------------------------
// Stage 5: out = feat + relu((trans - mu[f]) * rs[f]), in place on d_out.
// ---------------------------------------------------------------------------
__global__ __launch_bounds__(256) void finalize_kernel(const float* __restrict__ feat,
                                                       const float* __restrict__ stats,
                                                       float* __restrict__ out) {
  const size_t total4 = (size_t)B_ * F_ * N_ / 4;
  for (size_t i4 = (size_t)blockIdx.x * 256 + threadIdx.x; i4 < total4;
       i4 += (size_t)gridDim.x * 256) {
    size_t i = i4 * 4;
    int f = (int)((i >> 11) & (F_ - 1));
    float mu = stats[f];
    float rs = stats[F_ + f];
    v4f t  = *(const v4f*)(out + i);
    v4f fv = *(const v4f*)(feat + i);
    v4f r;
#pragma unroll
    for (int j = 0; j < 4; ++j) {
      float x = (t[j] - mu) * rs;
      r[j] = fv[j] + (x > 0.0f ? x : 0.0f);
    }
    *(v4f*)(out + i) = r;
  }
}

extern "C" void kernel_launch(void* const* d_in, const int* in_sizes, int n_in,
                              void* d_out, int out_size, void* d_ws, size_t ws_size,
                              hipStream_t stream) {
  const float* xyz   = (const float*)d_in[0];
  const float* feat  = (const float*)d_in[1];
  const float* W     = (const float*)d_in[2];
  const float* b_lin = (const float*)d_in[3];
  float* out = (float*)d_out;

  char* ws = (char*)d_ws;
  int*   idx   = (int*)ws;                         // 256 KB
  float* stats = (float*)(ws + (256 << 10));       // 16 KB
  float* lap   = (float*)(ws + (512 << 10));       // 64 MB

  knn_kernel<<<32, 256, 0, stream>>>(xyz, idx);
  lap_kernel<<<1024, 256, 0, stream>>>(feat, idx, lap);
  dim3 g(32, 16, 4);                               // (m tiles, f tiles, batch)
  gemm_kernel<<<g, 256, 0, stream>>>(lap, W, b_lin, out);
  stats_kernel<<<2048, 256, 0, stream>>>(out, stats);
  finalize_kernel<<<16384, 256, 0, stream>>>(feat, stats, out);
}